// REANN_88596585382861
// MI455X (gfx1250) — compile-verified
//
#include <hip/hip_runtime.h>

#define NTA   100000
#define NN    3000000
#define NS    4
#define NMAX  8
#define NO    4
#define H1D   4
#define H2D   9
#define PI_F  3.14159265358979323846f
#define RCUT_F 6.0f

typedef __attribute__((ext_vector_type(2))) float v2f;
typedef __attribute__((ext_vector_type(8))) float v8f;

// ---------------------------------------------------------------------------
// Edge scatter: per edge, compute fcut * angle[o] * radial[n] * Csn[j][n] and
// atomically accumulate into bnl[i][o][n]  ([NTA,4,8], L2-resident, 12.8 MB).
// ---------------------------------------------------------------------------
__global__ __launch_bounds__(256) void edge_scatter_kernel(
    const int* __restrict__ iidx, const int* __restrict__ jidx,
    const int* __restrict__ jsym, const float* __restrict__ dist,
    const float* __restrict__ disp, const float* __restrict__ alpha,
    const float* __restrict__ rs, const float* __restrict__ species_params,
    const float* __restrict__ csn,   // nullptr => pass 0: Csn[j] == species_params[jsym]
    float* __restrict__ bnl)
{
    int e = blockIdx.x * blockDim.x + threadIdx.x;
    if (e >= NN) return;
    int i = iidx[e];
    int j = jidx[e];
    int s = jsym[e];
    float d  = dist[e];
    float cc = __cosf(d * (PI_F / RCUT_F));
    float fc = 0.25f * (cc + 1.0f) * (cc + 1.0f);
    float dx = disp[3*e+0], dy = disp[3*e+1], dz = disp[3*e+2];

    const float* al = alpha + s * NMAX;
    const float* r0 = rs    + s * NMAX;
    const float* cj = csn ? (csn + (size_t)j * NMAX) : (species_params + s * NMAX);

    float* out = bnl + (size_t)i * (NO * NMAX);
    #pragma unroll
    for (int n = 0; n < NMAX; ++n) {
        float dr = d - r0[n];
        float g  = fc * __expf(al[n] * dr * dr) * cj[n];
        atomicAdd(out + n,            g);       // o=0: angle 1
        atomicAdd(out + NMAX + n,     g * dx);  // o=1: x
        atomicAdd(out + 2*NMAX + n,   g * dy);  // o=2: y
        atomicAdd(out + 3*NMAX + n,   g * dz);  // o=3: z
    }
}

// ---------------------------------------------------------------------------
// Density: per 16-atom tile, t[m,o] = sum_n bnl[m,l,n] * W[l][n,o] via
// V_WMMA_F32_16X16X4_F32 (two chained K=4 WMMAs per l), rho = sum_l t*t.
// A layout (16x4 f32): lanes 0-15 M=0..15 {K0,K1}; lanes 16-31 M=0..15 {K2,K3}.
// C/D layout: VGPR v: lanes 0-15 -> (M=v, N=lane), lanes 16-31 -> (M=v+8, N=lane-16).
// ---------------------------------------------------------------------------
#define DWAVES 4
#define NTILES (NTA / 16)   // 6250 exactly

__global__ __launch_bounds__(32 * DWAVES) void density_wmma_kernel(
    const float* __restrict__ bnl,   // [NTA,4,8]
    const float* __restrict__ W,     // orbital_params for this loop iter: [LMAX=2, 8, 4]
    float* __restrict__ rho)         // [NTA,4]
{
    int lane = threadIdx.x & 31;
    int wave = threadIdx.x >> 5;
    int tile = blockIdx.x * DWAVES + wave;
    if (tile >= NTILES) return;          // wave-uniform: EXEC all-1 below

    int m16 = lane & 15;
    int hi  = lane >> 4;                 // 0: K0/K1, M=v ; 1: K2/K3, M=v+8
    int col = m16;                       // B/D column (orbital o); only 0..3 real
    int ccol = (col < NO) ? col : 0;     // clamp to stay in-bounds
    float bmask = (col < NO) ? 1.0f : 0.0f;

    v8f acc = {};
    #pragma unroll
    for (int l = 0; l < NO; ++l) {
        const float* Wl = W + ((l == 0) ? 0 : 1) * (NMAX * NO);  // oidx = [0,1,1,1]
        v2f b0, b1;
        b0.x = Wl[(2*hi + 0) * NO + ccol] * bmask;   // K = 0 + 2*hi
        b0.y = Wl[(2*hi + 1) * NO + ccol] * bmask;   // K = 1 + 2*hi
        b1.x = Wl[(4 + 2*hi + 0) * NO + ccol] * bmask; // K = 4 + 2*hi
        b1.y = Wl[(4 + 2*hi + 1) * NO + ccol] * bmask; // K = 5 + 2*hi

        const float* arow = bnl + (size_t)(tile * 16 + m16) * (NO * NMAX) + l * NMAX + 2*hi;
        v2f a0, a1;
        a0.x = arow[0]; a0.y = arow[1];  // K chunk 0..3
        a1.x = arow[4]; a1.y = arow[5];  // K chunk 4..7

        v8f c = {};
        c = __builtin_amdgcn_wmma_f32_16x16x4_f32(false, a0, false, b0, (short)0, c, false, false);
        c = __builtin_amdgcn_wmma_f32_16x16x4_f32(false, a1, false, b1, (short)0, c, false, false);
        acc += c * c;
    }

    if (col < NO) {
        #pragma unroll
        for (int v = 0; v < 8; ++v) {
            int m = hi * 8 + v;
            rho[(size_t)(tile * 16 + m) * NO + col] = acc[v];
        }
    }
}

// ---------------------------------------------------------------------------
// Per-atom MLP with per-species weights: rho[4] -> silu(4) -> silu(9) -> 8,
// Csn1 = species_params[symbol] + coeff.
// ---------------------------------------------------------------------------
__global__ __launch_bounds__(256) void mlp_kernel(
    const int* __restrict__ symbols, const float* __restrict__ rho,
    const float* __restrict__ w1, const float* __restrict__ b1,
    const float* __restrict__ w2, const float* __restrict__ b2,
    const float* __restrict__ w3, const float* __restrict__ b3,
    const float* __restrict__ species_params,
    float* __restrict__ csn1)
{
    int a = blockIdx.x * blockDim.x + threadIdx.x;
    if (a >= NTA) return;
    int s = symbols[a];

    float r[NO];
    #pragma unroll
    for (int o = 0; o < NO; ++o) r[o] = rho[(size_t)a * NO + o];

    float h1v[H1D];
    #pragma unroll
    for (int h = 0; h < H1D; ++h) {
        float z = b1[s * H1D + h];
        #pragma unroll
        for (int o = 0; o < NO; ++o) z += r[o] * w1[(s * NO + o) * H1D + h];
        h1v[h] = z / (1.0f + __expf(-z));
    }
    float h2v[H2D];
    #pragma unroll
    for (int g = 0; g < H2D; ++g) {
        float z = b2[s * H2D + g];
        #pragma unroll
        for (int h = 0; h < H1D; ++h) z += h1v[h] * w2[(s * H1D + h) * H2D + g];
        h2v[g] = z / (1.0f + __expf(-z));
    }
    #pragma unroll
    for (int m = 0; m < NMAX; ++m) {
        float z = b3[s * NMAX + m];
        #pragma unroll
        for (int g = 0; g < H2D; ++g) z += h2v[g] * w3[(s * H2D + g) * NMAX + m];
        csn1[(size_t)a * NMAX + m] = species_params[s * NMAX + m] + z;
    }
}

// ---------------------------------------------------------------------------
extern "C" void kernel_launch(void* const* d_in, const int* in_sizes, int n_in,
                              void* d_out, int out_size, void* d_ws, size_t ws_size,
                              hipStream_t stream)
{
    const int*   symbols        = (const int*)  d_in[0];
    const int*   iidx           = (const int*)  d_in[1];
    const int*   jidx           = (const int*)  d_in[2];
    const int*   jsym           = (const int*)  d_in[3];
    const float* disp           = (const float*)d_in[4];
    const float* dist           = (const float*)d_in[5];
    const float* alpha          = (const float*)d_in[6];
    const float* rs             = (const float*)d_in[7];
    const float* species_params = (const float*)d_in[8];
    const float* orbital_params = (const float*)d_in[9];   // [2,2,8,4]
    const float* w1             = (const float*)d_in[10];
    const float* b1             = (const float*)d_in[11];
    const float* w2             = (const float*)d_in[12];
    const float* b2             = (const float*)d_in[13];
    const float* w3             = (const float*)d_in[14];
    const float* b3             = (const float*)d_in[15];
    float* out = (float*)d_out;

    float* bnl  = (float*)d_ws;                 // NTA*32 floats = 12.8 MB
    float* csn1 = bnl  + (size_t)NTA * 32;      // NTA*8  floats =  3.2 MB
    float* rho0 = csn1 + (size_t)NTA * 8;       // NTA*4  floats =  1.6 MB
    const size_t bnlBytes = (size_t)NTA * 32 * sizeof(float);

    dim3 eb(256), eg((NN + 255) / 256);
    dim3 db(32 * DWAVES), dg((NTILES + DWAVES - 1) / DWAVES);
    dim3 mb(256), mg((NTA + 255) / 256);

    const int loopStride = 2 * NMAX * NO;       // LMAX*NMAX*NO = 64

    // Pass 0
    hipMemsetAsync(bnl, 0, bnlBytes, stream);
    edge_scatter_kernel<<<eg, eb, 0, stream>>>(iidx, jidx, jsym, dist, disp,
                                               alpha, rs, species_params,
                                               nullptr, bnl);
    density_wmma_kernel<<<dg, db, 0, stream>>>(bnl, orbital_params, rho0);
    mlp_kernel<<<mg, mb, 0, stream>>>(symbols, rho0, w1, b1, w2, b2, w3, b3,
                                      species_params, csn1);
    // Pass 1
    hipMemsetAsync(bnl, 0, bnlBytes, stream);
    edge_scatter_kernel<<<eg, eb, 0, stream>>>(iidx, jidx, jsym, dist, disp,
                                               alpha, rs, species_params,
                                               csn1, bnl);
    density_wmma_kernel<<<dg, db, 0, stream>>>(bnl, orbital_params + loopStride, out);
}